// SpinHamiltonian_83665962926683
// MI455X (gfx1250) — compile-verified
//
#include <hip/hip_runtime.h>
#include <hip/hip_bf16.h>

// Heisenberg-spin energy on 128x128 periodic lattice, one sample per block.
// u = (cos p, sin p cos a, sin p sin a); E = -sum_bonds u . u'.
// Pipeline: async-DMA raw angles -> LDS (double buffered, non-temporal),
// sincos transform -> xyz in LDS, bond accumulation from LDS, reduction.

#define SPIN_L  128
#define SPIN_V  (SPIN_L * SPIN_L)      // 16384 sites
#define SPIN_CH 2048                   // sites per async chunk
#define SPIN_NCH (SPIN_V / SPIN_CH)    // 8 chunks

// one global_load_async_to_lds_b128 per lane: 4 floats global -> LDS.
// Data is streamed exactly once -> non-temporal hint (TH=NT) to spare L2.
__device__ __forceinline__ void async_copy16(unsigned lds_byte_off, const void* gptr) {
    unsigned long long ga = (unsigned long long)gptr;
    asm volatile("global_load_async_to_lds_b128 %0, %1, off th:TH_LOAD_NT"
                 :: "v"(lds_byte_off), "v"(ga) : "memory");
}

__device__ __forceinline__ void wait_async_le1() {
    asm volatile("s_wait_asynccnt 1" ::: "memory");
}
__device__ __forceinline__ void wait_async_le0() {
    asm volatile("s_wait_asynccnt 0" ::: "memory");
}

// Issue the async loads for chunk c (2048 polar + 2048 azimuth floats).
// Threads 0..511 carry polar, 512..1023 carry azimuth; one b128 per lane.
__device__ __forceinline__ void issue_chunk(const float* polar, const float* az,
                                            unsigned raw_base_bytes, int c, int tid) {
    const int  grp  = tid & 511;            // 512 lanes x 4 floats = 2048 floats
    const bool isAz = tid >= 512;
    const float* g  = (isAz ? az : polar) + c * SPIN_CH + grp * 4;
    unsigned lds = raw_base_bytes
                 + ((c & 1) ? (unsigned)(SPIN_CH * 2 * 4) : 0u)   // double buffer
                 + (isAz ? (unsigned)(SPIN_CH * 4) : 0u)          // az half
                 + (unsigned)(grp * 16);
    async_copy16(lds, (const void*)g);
}

__global__ __launch_bounds__(1024)
void SpinHamiltonian_kernel(const float* __restrict__ state, float* __restrict__ out) {
    extern __shared__ float smem[];
    float* X   = smem;                  // V floats
    float* Y   = smem + SPIN_V;         // V floats
    float* Z   = smem + 2 * SPIN_V;     // V floats
    float* RAW = smem + 3 * SPIN_V;     // 2 x 4096 floats (polar|az per buffer)

    const int tid    = (int)threadIdx.x;
    const int sample = (int)blockIdx.x;
    const float* polar = state + (size_t)sample * (size_t)(2 * SPIN_V);
    const float* az    = polar + SPIN_V;

    // LDS byte offset of RAW (generic LDS pointer low 32 bits == wave LDS offset).
    const unsigned raw_base = (unsigned)(unsigned long long)(void*)RAW;

    // ---- Phase 1: DMA + transform, double buffered -------------------------
    issue_chunk(polar, az, raw_base, 0, tid);
    for (int c = 0; c < SPIN_NCH; ++c) {
        if (c + 1 < SPIN_NCH) {
            issue_chunk(polar, az, raw_base, c + 1, tid);  // prefetch next
            wait_async_le1();                              // chunk c complete
        } else {
            wait_async_le0();
        }
        __syncthreads();                                   // all waves' DMA visible

        float* rb = RAW + ((c & 1) ? (SPIN_CH * 2) : 0);
        #pragma unroll
        for (int k = 0; k < 2; ++k) {
            const int l = tid + k * 1024;                  // 2048 sites / 1024 thr
            const float p = rb[l];
            const float a = rb[SPIN_CH + l];
            float sp, cp, sa, ca;
            __sincosf(p, &sp, &cp);
            __sincosf(a, &sa, &ca);
            const int s = c * SPIN_CH + l;
            X[s] = cp;
            Y[s] = sp * ca;
            Z[s] = sp * sa;
        }
        __syncthreads();                                   // buffer free for reuse
    }

    // ---- Phase 2: bond accumulation (down + right neighbors, periodic) -----
    float acc = 0.0f;
    #pragma unroll
    for (int j = 0; j < SPIN_V / 1024; ++j) {
        const int s  = tid + j * 1024;
        const int sd = (s + SPIN_L) & (SPIN_V - 1);                    // roll axis0
        const int sr = (s & ~(SPIN_L - 1)) | ((s + 1) & (SPIN_L - 1)); // roll axis1
        acc += X[s] * (X[sd] + X[sr])
             + Y[s] * (Y[sd] + Y[sr])
             + Z[s] * (Z[sd] + Z[sr]);
    }

    // ---- Phase 3: reduction (wave32 shuffle, then cross-wave via LDS) ------
    for (int o = 16; o; o >>= 1) acc += __shfl_down(acc, o, 32);
    float* red = RAW;                    // raw buffers are dead now
    const int wave = tid >> 5;
    const int lane = tid & 31;
    __syncthreads();
    if (lane == 0) red[wave] = acc;
    __syncthreads();
    if (wave == 0) {
        float v = red[lane];
        for (int o = 16; o; o >>= 1) v += __shfl_down(v, o, 32);
        if (lane == 0) out[sample] = -v;   // BETA = 1.0
    }
}

extern "C" void kernel_launch(void* const* d_in, const int* in_sizes, int n_in,
                              void* d_out, int out_size, void* d_ws, size_t ws_size,
                              hipStream_t stream) {
    (void)n_in; (void)out_size; (void)d_ws; (void)ws_size;
    const float* state = (const float*)d_in[0];
    // d_in[1] (shift) is unused: neighbor indices are analytic for roll(-1).
    const int n_sample = in_sizes[0] / (2 * SPIN_V);
    const size_t shmem = (size_t)(3 * SPIN_V + 2 * 2 * SPIN_CH) * sizeof(float); // 224 KB
    SpinHamiltonian_kernel<<<dim3((unsigned)n_sample), dim3(1024), shmem, stream>>>(
        state, (float*)d_out);
}